// BinarySplitDecoder_39118562132378
// MI455X (gfx1250) — compile-verified
//
#include <hip/hip_runtime.h>
#include <cstdint>

// BinarySplitDecoder: out[b,leaf] = prod_d ( bit(9-d,leaf) ? 1-x[b,node] : x[b,node] ),
// node = 2^d-1 + (leaf >> (10-d)).  Memory-bound: ~0.54 GB traffic -> ~23us @ 23.3 TB/s.
// CDNA5 path: async global->LDS row staging (ASYNCcnt) with a swizzled LDS layout that
// makes every tree level aligned + bank-conflict-free, double-buffered across rows.
// Streaming data (read-once input, write-once output) uses NT temporal hints since the
// 536 MB stream exceeds the 192 MB L2.

#define WAVES_PER_BLOCK 4
#define ROWS_PER_WAVE   8
#define NIN             1023
#define NOUT            1024
#define BUF_DW          1088   // swizzled row buffer (dwords); max map index = 1083

typedef float v2f __attribute__((ext_vector_type(2)));
typedef float v4f __attribute__((ext_vector_type(4)));

// LDS dword of input element k:  map(k) = (k+1) + 4*((k+1)>>6)
//  - "+1" shift: level base 2^d-1 -> 2^d  => b64/b128 DS loads naturally aligned
//  - 4-dword pad per 64: de-conflicts stride-8/16 bottom-level b128 reads
// dword 0 of each buffer is never a map target -> used as scratch for the tail lane.

__device__ __forceinline__ void async_ld_b32(uint32_t lds_byte, uint32_t gl_byte,
                                             const float* sbase) {
  // GVS mode: mem = SGPR64 + VGPR32 offset; VDST VGPR = LDS byte address.
  // Input is read exactly once -> non-temporal load hint.
  asm volatile("global_load_async_to_lds_b32 %0, %1, %2 th:TH_LOAD_NT"
               :: "v"(lds_byte), "v"(gl_byte), "s"(sbase)
               : "memory");
}

#define WAIT_ASYNCCNT(n) asm volatile("s_wait_asynccnt %0" :: "n"(n) : "memory")

__device__ __forceinline__ void issue_row(const float* __restrict__ x, uint32_t row,
                                          uint32_t lane, float* buf) {
  uint32_t bufbase = (uint32_t)(uintptr_t)buf;   // low 32 bits of LDS generic ptr = LDS byte offset
  uint32_t rowbyte = row * (uint32_t)(NIN * 4);
#pragma unroll
  for (int c = 0; c < 32; ++c) {
    uint32_t t    = (uint32_t)(32 * c) + lane + 1u;          // 1..1024
    uint32_t tail = (t > (uint32_t)NIN) ? 1u : 0u;           // only c==31, lane==31
    uint32_t k    = tail ? (uint32_t)(NIN - 1) : (t - 1u);   // clamp global read in-bounds
    uint32_t ldsb = tail ? bufbase : (bufbase + 4u * t + 16u * (t >> 6));
    async_ld_b32(ldsb, rowbyte + 4u * k, x);
  }
}

__device__ __forceinline__ void compute_row(const float* __restrict__ buf, uint32_t lane,
                                            float* __restrict__ outrow) {
  // ---- prefix: depths 0..4 (node = lane >> (5-d); sel bit = lane bit (4-d)) ----
  float P = 1.0f;
#pragma unroll
  for (int d = 0; d < 5; ++d) {
    uint32_t node = lane >> (5 - d);
    uint32_t k    = (1u << d) - 1u + node;        // <= 30  -> map(k) = k+1
    float a = buf[k + 1u];
    P *= ((lane >> (4 - d)) & 1u) ? (1.0f - a) : a;
  }

  // ---- depth 5: lds dword 32+lane ----
  float a5 = buf[32 + lane];
  float p5[2];
  p5[0] = P * a5;  p5[1] = P - p5[0];

  // ---- depth 6: b64 @ dword 68+2*lane (8B aligned, conflict-free) ----
  const v2f a6 = *(const v2f*)&buf[68 + 2 * lane];
  float p6[4];
  p6[0] = p5[0] * a6.x;  p6[1] = p5[0] - p6[0];
  p6[2] = p5[1] * a6.y;  p6[3] = p5[1] - p6[2];

  // ---- depth 7: b128 @ dword 136 + 4*lane + 4*(lane>=16) ----
  const v4f a7 = *(const v4f*)&buf[136 + 4 * lane + ((lane >> 4) << 2)];
  float p7[8];
  p7[0] = p6[0] * a7.x;  p7[1] = p6[0] - p7[0];
  p7[2] = p6[1] * a7.y;  p7[3] = p6[1] - p7[2];
  p7[4] = p6[2] * a7.z;  p7[5] = p6[2] - p7[4];
  p7[6] = p6[3] * a7.w;  p7[7] = p6[3] - p7[6];

  // ---- depth 8: 2x b128 @ dword 272 + 8*lane + 4*(lane>>3) ----
  uint32_t g8 = 272 + 8 * lane + ((lane >> 3) << 2);
  const v4f a8a = *(const v4f*)&buf[g8];
  const v4f a8b = *(const v4f*)&buf[g8 + 4];
  float p8[16];
  p8[ 0] = p7[0] * a8a.x;  p8[ 1] = p7[0] - p8[ 0];
  p8[ 2] = p7[1] * a8a.y;  p8[ 3] = p7[1] - p8[ 2];
  p8[ 4] = p7[2] * a8a.z;  p8[ 5] = p7[2] - p8[ 4];
  p8[ 6] = p7[3] * a8a.w;  p8[ 7] = p7[3] - p8[ 6];
  p8[ 8] = p7[4] * a8b.x;  p8[ 9] = p7[4] - p8[ 8];
  p8[10] = p7[5] * a8b.y;  p8[11] = p7[5] - p8[10];
  p8[12] = p7[6] * a8b.z;  p8[13] = p7[6] - p8[12];
  p8[14] = p7[7] * a8b.w;  p8[15] = p7[7] - p8[14];

  // ---- depth 9 + store: 4x b128 @ dword 544 + 16*lane + 4*(lane>>2) ----
  uint32_t g9 = 544 + 16 * lane + ((lane >> 2) << 2);
  v4f* outv = (v4f*)(outrow + 32 * lane);   // 16B aligned (row stride 4096B)
#pragma unroll
  for (int q = 0; q < 4; ++q) {
    const v4f a9 = *(const v4f*)&buf[g9 + 4 * q];
    v4f o0, o1;
    float b0 = p8[4 * q + 0] * a9.x;  o0.x = b0;  o0.y = p8[4 * q + 0] - b0;
    float b1 = p8[4 * q + 1] * a9.y;  o0.z = b1;  o0.w = p8[4 * q + 1] - b1;
    float b2 = p8[4 * q + 2] * a9.z;  o1.x = b2;  o1.y = p8[4 * q + 2] - b2;
    float b3 = p8[4 * q + 3] * a9.w;  o1.z = b3;  o1.w = p8[4 * q + 3] - b3;
    // write-once stream: NT store hint; each 128B line is fully covered by this lane's
    // 8 consecutive stores, so GL2 write-combines full lines before NT eviction.
    __builtin_nontemporal_store(o0, &outv[2 * q]);
    __builtin_nontemporal_store(o1, &outv[2 * q + 1]);
  }
}

__global__ __launch_bounds__(WAVES_PER_BLOCK * 32)
void BinarySplitDecoder_kernel(const float* __restrict__ x, float* __restrict__ out, int B) {
  __shared__ float lds[WAVES_PER_BLOCK][2][BUF_DW];   // 34,816 B / block
  const uint32_t tid  = threadIdx.x;
  const uint32_t w    = tid >> 5;
  const uint32_t lane = tid & 31u;

  uint32_t row0 = (blockIdx.x * WAVES_PER_BLOCK + w) * ROWS_PER_WAVE;
  if (row0 >= (uint32_t)B) return;

  issue_row(x, row0, lane, &lds[w][0][0]);            // prime the pipeline (32 async ops)

#pragma unroll
  for (int i = 0; i < ROWS_PER_WAVE; ++i) {
    uint32_t row = row0 + (uint32_t)i;
    if (row >= (uint32_t)B) break;
    bool havenext = (i + 1 < ROWS_PER_WAVE) && (row + 1u < (uint32_t)B);
    if (havenext) {
      issue_row(x, row + 1u, lane, &lds[w][(i + 1) & 1][0]);  // prefetch next row
      WAIT_ASYNCCNT(32);                                      // in-order: row i complete
    } else {
      WAIT_ASYNCCNT(0);
    }
    compute_row(&lds[w][i & 1][0], lane, out + (size_t)row * NOUT);
  }
}

extern "C" void kernel_launch(void* const* d_in, const int* in_sizes, int n_in,
                              void* d_out, int out_size, void* d_ws, size_t ws_size,
                              hipStream_t stream) {
  const float* x = (const float*)d_in[0];
  float* out     = (float*)d_out;
  int B = in_sizes[0] / NIN;                                  // 65536
  int waves  = (B + ROWS_PER_WAVE - 1) / ROWS_PER_WAVE;       // 8192
  int blocks = (waves + WAVES_PER_BLOCK - 1) / WAVES_PER_BLOCK; // 2048
  BinarySplitDecoder_kernel<<<blocks, WAVES_PER_BLOCK * 32, 0, stream>>>(x, out, B);
}